// IRadon_42374147343115
// MI455X (gfx1250) — compile-verified
//
#include <hip/hip_runtime.h>
#include <math.h>

typedef __attribute__((ext_vector_type(2))) float v2f;
typedef __attribute__((ext_vector_type(8))) float v8f;

#define NB    8      // batches
#define DETN  512    // detector cells (= output image side)
#define NANG  180    // angles
// angle tiles padded to 192 (12 tiles of 16)

// ---------------------------------------------------------------------------
// Phase 1: ramp filter as exact f32 GEMM via V_WMMA_F32_16X16X4_F32
//   xsT[b][a][d] = sum_k H[d,k] * x[b][0][k][a],  H[d,k] = h(|d-k|)
//   h(0)=0.5, h(m odd)=-2/(pi*m)^2, h(m even)=0
// One wave -> one 32(M=d) x 16(N=a) tile: two independent WMMA accumulator
// chains sharing each B fragment (2x matrix-pipe ILP, half the B traffic).
// ---------------------------------------------------------------------------
__global__ __launch_bounds__(256) void ramp_gemm_wmma(const float* __restrict__ x,
                                                      float* __restrict__ xsT) {
    __shared__ float htab[DETN];
    const int tid = threadIdx.x;
    for (int i = tid; i < DETN; i += 256) {
        float v = 0.0f;
        if (i == 0) {
            v = 0.5f;
        } else if (i & 1) {
            double di = (double)i;
            v = (float)(-2.0 / (M_PI * M_PI * di * di));
        }
        htab[i] = v;
    }
    __syncthreads();

    const int wave = tid >> 5;
    const int lane = tid & 31;
    const int tile = blockIdx.x * 8 + wave;       // 1536 tiles total
    const int batch = tile / (16 * 12);
    const int rem   = tile % (16 * 12);
    const int d0 = (rem / 12) * 32;               // 32-row M tile
    const int a0 = (rem % 12) * 16;

    const int ln = lane & 15;                     // M (A) / N (B,D) index
    const int hi = lane >> 4;                     // half-wave selector
    const int dm0 = d0 + ln;                      // A row, lower 16x16 tile
    const int dm1 = d0 + 16 + ln;                 // A row, upper 16x16 tile
    const int arow = a0 + ln;                     // B/D column (angle)
    const bool avalid = arow < NANG;
    const int acl = avalid ? arow : (NANG - 1);   // clamped (branchless guard)

    const float* xb = x + (size_t)batch * DETN * NANG;

    v8f acc0 = {};
    v8f acc1 = {};
    #pragma unroll 4
    for (int k0 = 0; k0 < DETN; k0 += 4) {
        const int ka = k0 + 2 * hi;               // this lane's first K
        // A fragments from LDS ramp table (exec stays all-ones: pure selects)
        int p0 = dm0 - ka;       p0 = p0 < 0 ? -p0 : p0;
        int p1 = dm0 - (ka + 1); p1 = p1 < 0 ? -p1 : p1;
        int q0 = dm1 - ka;       q0 = q0 < 0 ? -q0 : q0;
        int q1 = dm1 - (ka + 1); q1 = q1 < 0 ? -q1 : q1;
        v2f afA, afB;
        afA.x = htab[p0];
        afA.y = htab[p1];
        afB.x = htab[q0];
        afB.y = htab[q1];
        // Shared B fragment from global (angle column clamped, zeroed by select)
        float b0 = xb[(size_t)ka * NANG + acl];
        float b1 = xb[(size_t)(ka + 1) * NANG + acl];
        v2f bf;
        bf.x = avalid ? b0 : 0.0f;
        bf.y = avalid ? b1 : 0.0f;
        // Two independent accumulator chains
        acc0 = __builtin_amdgcn_wmma_f32_16x16x4_f32(false, afA, false, bf,
                                                     (short)0, acc0, false, false);
        acc1 = __builtin_amdgcn_wmma_f32_16x16x4_f32(false, afB, false, bf,
                                                     (short)0, acc1, false, false);
    }

    // D layout: VGPR r -> row m = r + 8*hi, column n = ln.
    // Store transposed: xsT[b][a][d], d contiguous.
    if (avalid) {
        float* outp = xsT + ((size_t)batch * NANG + (size_t)arow) * DETN + d0 + 8 * hi;
        #pragma unroll
        for (int r = 0; r < 8; ++r) outp[r] = acc0[r];
        #pragma unroll
        for (int r = 0; r < 8; ++r) outp[16 + r] = acc1[r];
    }
}

// ---------------------------------------------------------------------------
// Phase 2: backprojection. Block = (b, image row y), 256 threads x 2 pixels.
// sin/cos for all 180 angles staged in LDS once per block. Filtered sinogram
// (2.95 MB) is L2-resident; gathers are unit-stride across lanes.
// ---------------------------------------------------------------------------
__global__ __launch_bounds__(256) void backproject(const float* __restrict__ xsT,
                                                   float* __restrict__ out) {
    __shared__ float csT[NANG];
    __shared__ float snT[NANG];
    const int tid = threadIdx.x;
    if (tid < NANG) {
        float th = (float)tid * 0.017453292519943295f;  // pi/180
        csT[tid] = cosf(th);
        snT[tid] = sinf(th);
    }
    __syncthreads();

    const int b = blockIdx.x >> 9;           // / 512
    const int y = blockIdx.x & 511;
    const float step = 2.0f / 511.0f;        // linspace(-1,1,512) increment
    const float yg  = -1.0f + step * (float)y;
    const int   x0i = tid;
    const int   x1i = tid + 256;
    const float xg0 = -1.0f + step * (float)x0i;
    const float xg1 = -1.0f + step * (float)x1i;

    const float* base = xsT + (size_t)b * NANG * DETN;
    float acc0 = 0.0f, acc1 = 0.0f;

    for (int a = 0; a < NANG; ++a) {
        const float c = csT[a];
        const float s = snT[a];
        const float* col = base + (size_t)a * DETN;
        const float ys = yg * s;

        {   // pixel x0
            float t  = xg0 * c - ys;
            float yi = (t + 1.0f) * 255.5f;   // * 0.5*(D-1)
            float fi = floorf(yi);
            int   i0 = (int)fi;
            float w  = yi - fi;
            int i0c = i0 < 0 ? 0 : (i0 > 511 ? 511 : i0);
            int i1  = i0 + 1;
            int i1c = i1 < 0 ? 0 : (i1 > 511 ? 511 : i1);
            float v0 = col[i0c]; if (i0 < 0 || i0 > 511) v0 = 0.0f;
            float v1 = col[i1c]; if (i1 < 0 || i1 > 511) v1 = 0.0f;
            acc0 += v0 * (1.0f - w) + v1 * w;
        }
        {   // pixel x1
            float t  = xg1 * c - ys;
            float yi = (t + 1.0f) * 255.5f;
            float fi = floorf(yi);
            int   i0 = (int)fi;
            float w  = yi - fi;
            int i0c = i0 < 0 ? 0 : (i0 > 511 ? 511 : i0);
            int i1  = i0 + 1;
            int i1c = i1 < 0 ? 0 : (i1 > 511 ? 511 : i1);
            float v0 = col[i0c]; if (i0 < 0 || i0 > 511) v0 = 0.0f;
            float v1 = col[i1c]; if (i1 < 0 || i1 > 511) v1 = 0.0f;
            acc1 += v0 * (1.0f - w) + v1 * w;
        }
    }

    const float scale = (float)(M_PI / (2.0 * (double)NANG));
    float* orow = out + ((size_t)b * DETN + (size_t)y) * DETN;
    float r0 = xg0 * xg0 + yg * yg;
    float r1 = xg1 * xg1 + yg * yg;
    orow[x0i] = (r0 <= 1.0f) ? acc0 * scale : 0.0f;
    orow[x1i] = (r1 <= 1.0f) ? acc1 * scale : 0.0f;
}

// ---------------------------------------------------------------------------
extern "C" void kernel_launch(void* const* d_in, const int* in_sizes, int n_in,
                              void* d_out, int out_size, void* d_ws, size_t ws_size,
                              hipStream_t stream) {
    (void)in_sizes; (void)n_in; (void)out_size; (void)ws_size;
    const float* x = (const float*)d_in[0];        // (8,1,512,180) f32
    float* xsT = (float*)d_ws;                     // (8,180,512) f32 = 2.95 MB
    float* out = (float*)d_out;                    // (8,1,512,512) f32

    // 1536 wave-tiles (32x16 each) / 8 waves per block
    ramp_gemm_wmma<<<192, 256, 0, stream>>>(x, xsT);
    // one block per (batch, image row)
    backproject<<<NB * DETN, 256, 0, stream>>>(xsT, out);
}